// OFDM_channel_21303037788504
// MI455X (gfx1250) — compile-verified
//
#include <hip/hip_runtime.h>
#include <math.h>

// Problem constants (from reference): N=1024, P=4, S=12, M=64, K=16, L=8
constexpr int kPairs = 4096;        // N*P independent (n,p) problems
constexpr int kS     = 12;          // payload symbols
constexpr int kM     = 64;          // subcarriers (FFT size)
constexpr int kK     = 16;          // cyclic prefix
constexpr int kL     = 8;           // channel taps
constexpr int kMK    = kM + kK;     // 80
constexpr int kNSym  = kS + 1;      // 13 symbols (pilot + payload)
constexpr int kBlock = 128;         // 4 wave32 waves per workgroup

// Output layout (floats, concatenated in return order)
constexpr size_t kIpOff  = 0;                                        // (N,P,1,M,2)
constexpr size_t kIsOff  = (size_t)kPairs * kM * 2;                  // 524288
constexpr size_t kHOff   = kIsOff + (size_t)kPairs * kS * kM * 2;    // 6815744
constexpr size_t kNpwOff = kHOff + (size_t)kPairs * kM * 2;          // 7340032

constexpr float kNoisePwr = 1.5625e-4f;          // 1/(64*10^(0.1*20))
constexpr float kNScale   = 0.008838834764831844f; // sqrt(noise_pwr/2)

typedef __attribute__((ext_vector_type(2))) float v2f;
typedef __attribute__((ext_vector_type(8))) float v8f;

// One complex (16x64)·(64x64) DFT matmul, all 4 waves cooperating; wave w owns
// output columns [16w,16w+16). B[k][j] = (ctab*cs) + i*(stab*ss) at index (k*j)&63:
//   IFFT: cs = ss = 1/64     FFT: cs = 1, ss = -1
__device__ __forceinline__ void dft16x64(const float* Ar, const float* Ai,
                                         const float* ctab, const float* stab,
                                         float cs, float ss, int lane, int w,
                                         v8f& accr, v8f& acci)
{
  const int half = lane >> 4;       // A/B fragment K-half select
  const int mrow = lane & 15;       // A fragment row
  const int jj   = w * 16 + (lane & 15);  // B fragment column (matmul N)
#pragma unroll
  for (int ks = 0; ks < 16; ++ks) {
    const int k0 = ks * 4 + half * 2;
    v2f ar, ai, br, bi, bin;
    ar.x = Ar[mrow * 64 + k0];  ar.y = Ar[mrow * 64 + k0 + 1];
    ai.x = Ai[mrow * 64 + k0];  ai.y = Ai[mrow * 64 + k0 + 1];
    const int i0 = (k0 * jj) & 63;
    const int i1 = ((k0 + 1) * jj) & 63;
    br.x  = ctab[i0] * cs;  br.y  = ctab[i1] * cs;
    bi.x  = stab[i0] * ss;  bi.y  = stab[i1] * ss;
    bin.x = -bi.x;          bin.y = -bi.y;
    // Cr += Ar*Br - Ai*Bi ; Ci += Ar*Bi + Ai*Br   (f32 WMMA: no A/B NEG, negate B in VALU)
    accr = __builtin_amdgcn_wmma_f32_16x16x4_f32(false, ar, false, br,  (short)0, accr, false, false);
    accr = __builtin_amdgcn_wmma_f32_16x16x4_f32(false, ai, false, bin, (short)0, accr, false, false);
    acci = __builtin_amdgcn_wmma_f32_16x16x4_f32(false, ar, false, bi,  (short)0, acci, false, false);
    acci = __builtin_amdgcn_wmma_f32_16x16x4_f32(false, ai, false, br,  (short)0, acci, false, false);
  }
}

__global__ void __launch_bounds__(kBlock)
ofdm_channel_kernel(const float* __restrict__ x,
                    const float* __restrict__ pilot_raw,
                    const float* __restrict__ cof_unit,
                    const float* __restrict__ noise_unit,
                    float* __restrict__ out)
{
  __shared__ float ctab[64], stab[64];
  __shared__ float Ar[16 * 64], Ai[16 * 64];      // matmul A staging (freq, then time)
  __shared__ float sig_r[kNSym * kMK], sig_i[kNSym * kMK];  // serialized TD signal w/ CP
  __shared__ float cof_r[kL], cof_i[kL];
  __shared__ float red[kBlock];

  const int pair = blockIdx.x;
  const int tid  = threadIdx.x;
  const int lane = tid & 31;
  const int w    = tid >> 5;

  const float* xb  = x          + (size_t)pair * kS * kM * 2;
  const float* prb = pilot_raw  + (size_t)pair * kM * 2;
  const float* cub = cof_unit   + (size_t)pair * kL * 2;
  const float* nzb = noise_unit + (size_t)pair * kNSym * kMK * 2;
  float* ipb = out + kIpOff + (size_t)pair * kM * 2;
  float* isb = out + kIsOff + (size_t)pair * kS * kM * 2;
  float* hb  = out + kHOff  + (size_t)pair * kM * 2;

  // --- twiddle tables: cos/sin(2*pi*t/64) ---
  if (tid < 64) {
    const float th = (2.0f * 3.14159265358979323846f / 64.0f) * (float)tid;
    ctab[tid] = cosf(th);
    stab[tid] = sinf(th);
  }
  // --- pilot power: mean of 128 squares (blockDim == 128 exactly) ---
  {
    const float v = prb[tid];
    red[tid] = v * v;
  }
  __syncthreads();
  for (int off = kBlock / 2; off > 0; off >>= 1) {
    if (tid < off) red[tid] += red[tid + off];
    __syncthreads();
  }
  const float alpha = sqrtf(0.5f / (red[0] * (1.0f / 128.0f)));  // sqrt(PWR/2)/sqrt(power)

  // --- channel taps: sqrt(prof[t]/(2*sum)) * cof_unit ---
  if (tid < kL) {
    float psum = 0.f;
#pragma unroll
    for (int t = 0; t < kL; ++t) psum += expf(-(float)t * 0.25f);
    const float sc = sqrtf(expf(-(float)tid * 0.25f) / psum * 0.5f);
    cof_r[tid] = sc * cub[2 * tid];
    cof_i[tid] = sc * cub[2 * tid + 1];
  }

  // --- A = freq-domain symbols: row 0 scaled pilot, rows 1..12 payload, 13..15 zero ---
  for (int idx = tid; idx < 16 * 64; idx += kBlock) {
    const int r = idx >> 6, k = idx & 63;
    float re = 0.f, im = 0.f;
    if (r == 0)      { re = alpha * prb[2 * k]; im = alpha * prb[2 * k + 1]; }
    else if (r <= kS){ const float* e = xb + (((size_t)(r - 1) * kM) + k) * 2; re = e[0]; im = e[1]; }
    Ar[idx] = re; Ai[idx] = im;
  }
  __syncthreads();

  // --- IFFT via WMMA: B[k][j] = (cos + i sin)(2pi kj/64) / 64 ---
  v8f tr = {}; v8f ti = {};
  dft16x64(Ar, Ai, ctab, stab, 1.0f / 64.0f, 1.0f / 64.0f, lane, w, tr, ti);

  // --- scatter time domain + cyclic prefix into serialized signal ---
  {
    const int half = lane >> 4;
    const int j = w * 16 + (lane & 15);
#pragma unroll
    for (int vv = 0; vv < 8; ++vv) {
      const int m = vv + half * 8;              // C layout: M = v + 8*(lane>=16)
      if (m < kNSym) {
        sig_r[m * kMK + kK + j] = tr[vv];
        sig_i[m * kMK + kK + j] = ti[vv];
        if (j >= kM - kK) {                     // CP = last K time samples
          sig_r[m * kMK + (j - (kM - kK))] = tr[vv];
          sig_i[m * kMK + (j - (kM - kK))] = ti[vv];
        }
      }
    }
  }
  __syncthreads();

  // --- H_true[k] = sum_t cof[t] * e^{-2pi i t k / 64} ---
  if (tid < kM) {
    float hr = 0.f, hi = 0.f;
#pragma unroll
    for (int t = 0; t < kL; ++t) {
      const float c = ctab[(t * tid) & 63], s = stab[(t * tid) & 63];
      const float a = cof_r[t], b = cof_i[t];
      hr += a * c + b * s;
      hi += b * c - a * s;
    }
    hb[2 * tid] = hr; hb[2 * tid + 1] = hi;
  }

  // --- 8-tap complex FIR + AWGN, CP removed -> reuse Ar/Ai as FFT input (rows 13..15 stay 0) ---
  for (int idx = tid; idx < kNSym * 64; idx += kBlock) {
    const int s = idx >> 6, j = idx & 63;
    const int i = s * kMK + kK + j;             // i >= 16, i - t >= 9: no boundary check
    float yr = 0.f, yi = 0.f;
#pragma unroll
    for (int t = 0; t < kL; ++t) {
      const float sr = sig_r[i - t], si = sig_i[i - t];
      const float a = cof_r[t], b = cof_i[t];
      yr += a * sr - b * si;
      yi += a * si + b * sr;
    }
    const float* nz = nzb + (size_t)(s * kMK + kK + j) * 2;
    Ar[idx] = yr + kNScale * nz[0];
    Ai[idx] = yi + kNScale * nz[1];
  }
  __syncthreads();

  // --- FFT via WMMA: B[j][k] = cos - i sin; write straight to global ---
  v8f fr = {}; v8f fi = {};
  dft16x64(Ar, Ai, ctab, stab, 1.0f, -1.0f, lane, w, fr, fi);
  {
    const int half = lane >> 4;
    const int j = w * 16 + (lane & 15);
#pragma unroll
    for (int vv = 0; vv < 8; ++vv) {
      const int m = vv + half * 8;
      if (m == 0) {
        ipb[2 * j] = fr[vv]; ipb[2 * j + 1] = fi[vv];
      } else if (m <= kS) {
        float* o = isb + ((size_t)(m - 1) * kM + j) * 2;
        o[0] = fr[vv]; o[1] = fi[vv];
      }
    }
  }

  if (pair == 0 && tid == 0) out[kNpwOff] = kNoisePwr;  // scalar output
}

extern "C" void kernel_launch(void* const* d_in, const int* in_sizes, int n_in,
                              void* d_out, int out_size, void* d_ws, size_t ws_size,
                              hipStream_t stream) {
  (void)in_sizes; (void)n_in; (void)out_size; (void)d_ws; (void)ws_size;
  const float* x  = (const float*)d_in[0];
  const float* pr = (const float*)d_in[1];
  const float* cu = (const float*)d_in[2];
  const float* nz = (const float*)d_in[3];
  float* out = (float*)d_out;
  ofdm_channel_kernel<<<kPairs, kBlock, 0, stream>>>(x, pr, cu, nz, out);
}